// Voxelization_5892695130408
// MI455X (gfx1250) — compile-verified
//
#include <hip/hip_runtime.h>
#include <hip/hip_bf16.h>
#include <stdint.h>
#include <stddef.h>

#define MAXP      35
#define MAXV      20000
#define GXD       1408
#define GYD       1600
#define GZD       40
#define IDX_CAP   64
#define TB        256

// ---------------------------------------------------------------------------
// voxel coordinate, bit-exact vs jnp.floor((p - lo)/vs).astype(int32)
// lo = [0,-40,-3], vs = [0.05,0.05,0.1]; IEEE f32 sub + correctly-rounded div
// ---------------------------------------------------------------------------
__device__ __forceinline__ void vox_coord(float px, float py, float pz,
                                          int& cx, int& cy, int& cz) {
  float fx = (px - 0.0f)    / 0.05f;
  float fy = (py - (-40.0f)) / 0.05f;
  float fz = (pz - (-3.0f))  / 0.1f;
  cx = (int)floorf(fx);
  cy = (int)floorf(fy);
  cz = (int)floorf(fz);
}

__device__ __forceinline__ bool vox_valid(int cx, int cy, int cz) {
  return (cx >= 0) & (cx < GXD) & (cy >= 0) & (cy < GYD) & (cz >= 0) & (cz < GZD);
}

// ---------------------------------------------------------------------------
// zero / init kernels
// ---------------------------------------------------------------------------
__global__ void k_zero_i32(int* p, size_t n) {
  size_t i = (size_t)blockIdx.x * blockDim.x + threadIdx.x;
  size_t stride = (size_t)gridDim.x * blockDim.x;
  for (; i < n; i += stride) p[i] = 0;
}

__global__ void k_init_table(int* keys, int* fidx, int* cnt, size_t cap) {
  size_t i = (size_t)blockIdx.x * blockDim.x + threadIdx.x;
  if (i < cap) {
    keys[i] = -1;
    fidx[i] = 0x7FFFFFFF;
  }
  if (i < MAXV) cnt[i] = 0;
}

// ---------------------------------------------------------------------------
// pass 1: hash-insert voxel id, atomicMin first original index
// hash table sized to sit in MI455X's 192MB L2
// ---------------------------------------------------------------------------
__global__ void k_build(const float4* __restrict__ pts, int* keys, int* fidx,
                        int* hpos, unsigned mask, int n) {
  int i = blockIdx.x * TB + threadIdx.x;
  if (i >= n) return;
  // CDNA5 speculative prefetch (global_prefetch_b8) on the point stream
  int pf = i + 4096; if (pf >= n) pf = n - 1;
  __builtin_prefetch(pts + pf, 0, 1);

  float4 p = pts[i];
  int cx, cy, cz;
  vox_coord(p.x, p.y, p.z, cx, cy, cz);
  if (!vox_valid(cx, cy, cz)) { hpos[i] = -1; return; }
  int flat = (cx * GYD + cy) * GZD + cz;

  unsigned h = ((unsigned)flat * 2654435761u) & mask;
  while (true) {
    int k = atomicCAS(&keys[h], -1, flat);
    if (k == -1 || k == flat) break;
    h = (h + 1u) & mask;
  }
  atomicMin(&fidx[h], i);
  hpos[i] = (int)h;
}

// ---------------------------------------------------------------------------
// pass 2a: per-block count of "winner" points (first point of their voxel)
// ---------------------------------------------------------------------------
__global__ void k_flagsum(const int* __restrict__ fidx, const int* __restrict__ hpos,
                          int* bsums, int n) {
  int i = blockIdx.x * TB + threadIdx.x;
  int t = threadIdx.x;
  int flag = 0;
  if (i < n) {
    int h = hpos[i];
    if (h >= 0 && fidx[h] == i) flag = 1;
  }
  __shared__ int sh[TB];
  sh[t] = flag;
  __syncthreads();
  for (int s = TB / 2; s > 0; s >>= 1) {
    if (t < s) sh[t] += sh[t + s];
    __syncthreads();
  }
  if (t == 0) bsums[blockIdx.x] = sh[0];
}

// ---------------------------------------------------------------------------
// pass 2b: single-block exclusive scan of block sums (in place); voxel_num
// ---------------------------------------------------------------------------
__global__ void k_scan(int* bsums, int nblk, int* voxnum) {
  __shared__ int sh[1024];
  __shared__ int carry;
  int t = threadIdx.x;
  if (t == 0) carry = 0;
  __syncthreads();
  for (int base = 0; base < nblk; base += 1024) {
    int idx = base + t;
    int v = (idx < nblk) ? bsums[idx] : 0;
    sh[t] = v;
    __syncthreads();
    for (int off = 1; off < 1024; off <<= 1) {
      int add = (t >= off) ? sh[t - off] : 0;
      __syncthreads();
      sh[t] += add;
      __syncthreads();
    }
    int incl = sh[t];
    int c0 = carry;
    if (idx < nblk) bsums[idx] = c0 + (incl - v);
    __syncthreads();
    if (t == 1023) carry = c0 + incl;   // chunk total
    __syncthreads();
  }
  if (t == 0) {
    int total = carry;
    *voxnum = total < MAXV ? total : MAXV;
  }
}

// ---------------------------------------------------------------------------
// pass 2c: winners compute their voxel slot (first-appearance order),
// publish slot as ~slot in fidx[h], and write coors (zyx) for slot<MAXV
// ---------------------------------------------------------------------------
__global__ void k_assign(const float4* __restrict__ pts, int* fidx,
                         const int* __restrict__ hpos, const int* __restrict__ boffs,
                         int* coors, int n) {
  int i = blockIdx.x * TB + threadIdx.x;
  int t = threadIdx.x;
  int flag = 0, h = -1;
  if (i < n) {
    h = hpos[i];
    if (h >= 0 && fidx[h] == i) flag = 1;
  }
  __shared__ int sh[TB];
  sh[t] = flag;
  __syncthreads();
  for (int off = 1; off < TB; off <<= 1) {
    int add = (t >= off) ? sh[t - off] : 0;
    __syncthreads();
    sh[t] += add;
    __syncthreads();
  }
  int excl = sh[t] - flag;
  if (flag) {
    int slot = boffs[blockIdx.x] + excl;
    fidx[h] = ~slot;          // negative-encoded; can never collide with a point index
    if (slot < MAXV) {
      float4 p = pts[i];
      int cx, cy, cz;
      vox_coord(p.x, p.y, p.z, cx, cy, cz);
      coors[slot * 3 + 0] = cz;
      coors[slot * 3 + 1] = cy;
      coors[slot * 3 + 2] = cx;
    }
  }
}

// ---------------------------------------------------------------------------
// pass 3: every point looks up its voxel slot, appends its original index
// ---------------------------------------------------------------------------
__global__ void k_count(const int* __restrict__ fidx, const int* __restrict__ hpos,
                        int* cnt, int* idxbuf, int n) {
  int i = blockIdx.x * TB + threadIdx.x;
  if (i >= n) return;
  int h = hpos[i];
  if (h < 0) return;
  int v = fidx[h];            // ~slot (written by the voxel's winner)
  if (v >= 0) return;         // defensive; every occupied slot has a winner
  int slot = ~v;
  if (slot < MAXV) {
    int pos = atomicAdd(&cnt[slot], 1);
    if (pos < IDX_CAP) idxbuf[slot * IDX_CAP + pos] = i;
  }
}

// ---------------------------------------------------------------------------
// pass 4: per kept voxel, order points by original index (rank = count of
// smaller indices; indices unique) and emit rows + num_points
// ---------------------------------------------------------------------------
__global__ void k_emit(const float4* __restrict__ pts, const int* __restrict__ cnt,
                       const int* __restrict__ idxbuf, float4* voxels, int* nump) {
  int s = blockIdx.x * TB + threadIdx.x;
  if (s >= MAXV) return;
  int n = cnt[s];
  int k = n < MAXP ? n : MAXP;
  nump[s] = k;
  int m = n < IDX_CAP ? n : IDX_CAP;
  const int* buf = idxbuf + (size_t)s * IDX_CAP;
  for (int j = 0; j < m; j++) {
    int e = buf[j];
    int r = 0;
    for (int jj = 0; jj < m; jj++) r += (buf[jj] < e);
    if (r < MAXP) voxels[(size_t)s * MAXP + r] = pts[e];
  }
}

// ---------------------------------------------------------------------------
// host launch
// ---------------------------------------------------------------------------
extern "C" void kernel_launch(void* const* d_in, const int* in_sizes, int n_in,
                              void* d_out, int out_size, void* d_ws, size_t ws_size,
                              hipStream_t stream) {
  const float4* pts = (const float4*)d_in[0];
  const int n = in_sizes[0] / 4;              // 4,000,000 points
  const int nblk = (n + TB - 1) / TB;

  // output layout (all 32-bit; ints stored as raw bits):
  // voxels [MAXV,MAXP,4] f32 | coors [MAXV,3] i32 | num_points [MAXV] i32 | voxel_num i32
  float*  outf   = (float*)d_out;
  float4* voxels = (float4*)outf;
  int*    coors  = (int*)(outf + (size_t)MAXV * MAXP * 4);
  int*    nump   = coors + (size_t)MAXV * 3;
  int*    voxnum = nump + MAXV;

  // hash capacity: prefer 2^24 (keys+vals = 134MB -> resident in 192MB L2)
  size_t cap = (size_t)1 << 24;
  while (cap > ((size_t)1 << 22)) {
    size_t need = (2 * cap + (size_t)n + (size_t)nblk + MAXV +
                   (size_t)MAXV * IDX_CAP + 64) * sizeof(int);
    if (need <= ws_size) break;
    cap >>= 1;
  }
  unsigned mask = (unsigned)(cap - 1);

  int* keys   = (int*)d_ws;
  int* fidx   = keys + cap;
  int* hpos   = fidx + cap;
  int* bsums  = hpos + n;
  int* cnt    = bsums + nblk;
  int* idxbuf = cnt + MAXV;

  k_zero_i32<<<4096, TB, 0, stream>>>((int*)d_out, (size_t)out_size);
  k_init_table<<<(int)((cap + TB - 1) / TB), TB, 0, stream>>>(keys, fidx, cnt, cap);
  k_build<<<nblk, TB, 0, stream>>>(pts, keys, fidx, hpos, mask, n);
  k_flagsum<<<nblk, TB, 0, stream>>>(fidx, hpos, bsums, n);
  k_scan<<<1, 1024, 0, stream>>>(bsums, nblk, voxnum);
  k_assign<<<nblk, TB, 0, stream>>>(pts, fidx, hpos, bsums, coors, n);
  k_count<<<nblk, TB, 0, stream>>>(fidx, hpos, cnt, idxbuf, n);
  k_emit<<<(MAXV + TB - 1) / TB, TB, 0, stream>>>(pts, cnt, idxbuf, voxels, nump);
}